// Zonotope_5669356831097
// MI455X (gfx1250) — compile-verified
//
#include <hip/hip_runtime.h>
#include <hip/hip_bf16.h>

typedef __attribute__((ext_vector_type(2))) float v2f;
typedef __attribute__((ext_vector_type(4))) float f4;
typedef __attribute__((ext_vector_type(8))) float v8f;

#define EPS_C   0.01f
#define MEAN_C  0.1307f
#define SIGMA_C 0.3081f

#define D_IN  1024
#define HID   4096
#define O_OUT 10
#define NE    1152   // padded eps-panel width: 1024 err + term + value + zero pad (multiple of 128)

// ---------------- gfx1250 async global->LDS helpers ----------------
__device__ __forceinline__ unsigned lds_off(const void* p) {
  // shared aperture: addr[31:0] is the LDS byte offset
  return (unsigned)(size_t)p;
}
__device__ __forceinline__ void async_b128(unsigned lds_addr, const float* gaddr) {
  asm volatile("global_load_async_to_lds_b128 %0, %1, off"
               :: "v"(lds_addr), "v"(gaddr) : "memory");
}
__device__ __forceinline__ void async_b32(unsigned lds_addr, const float* gaddr) {
  asm volatile("global_load_async_to_lds_b32 %0, %1, off"
               :: "v"(lds_addr), "v"(gaddr) : "memory");
}
__device__ __forceinline__ void wait_async0() {
  asm volatile("s_wait_asynccnt 0x0" ::: "memory");
}

// ---------------- Kernel A: initial zonotope build + normalize ----------------
__global__ void zono_init(const float* __restrict__ inp,
                          float* __restrict__ v0, float* __restrict__ d0) {
  int t = blockIdx.x * blockDim.x + threadIdx.x;
  if (t < D_IN) {
    float x  = inp[t];
    float up = fminf(x + EPS_C, 1.0f);
    float lo = fmaxf(x - EPS_C, 0.0f);
    float val = up + lo;
    v0[t] = (val - MEAN_C) / SIGMA_C;
    d0[t] = up / SIGMA_C;             // (values - lower)/sigma = upper/sigma (diag radius)
  }
}

// ------------- Kernel B: layer1 (diag eps => column scaling) + relu-zono -------------
__global__ __launch_bounds__(256) void layer1_relu(
    const float* __restrict__ W1, const float* __restrict__ b1,
    const float* __restrict__ v0, const float* __restrict__ d0,
    float* __restrict__ E1) {
  int wave = threadIdx.x >> 5;
  int lane = threadIdx.x & 31;
  int row  = blockIdx.x * 8 + wave;

  const float* wrow = W1 + (size_t)row * D_IN;
  float dv = 0.f, dr = 0.f;
  for (int j = lane; j < D_IN; j += 32) {
    float w = wrow[j];
    dv += w * v0[j];
    dr += fabsf(w) * d0[j];           // d0 >= 0, so |w*d| = |w|*d
  }
  for (int off = 16; off; off >>= 1) {
    dv += __shfl_xor(dv, off, 32);
    dr += __shfl_xor(dr, off, 32);
  }
  float v1 = dv + b1[row];
  float r  = dr;
  float u  = v1 + r, l = v1 - r;
  float denom = u - l;
  float slope = u / ((denom == 0.f) ? 1.f : denom);
  float term  = (1.f - slope) * u * 0.5f;
  bool crossing = (u > 0.f) && (l < 0.f);
  bool dead     = (u <= 0.f);
  float s  = dead ? 0.f : (crossing ? slope : 1.f);
  float tc = dead ? 0.f : (crossing ? term  : 0.f);
  float vn = dead ? 0.f : (crossing ? (slope * v1 + term) : v1);

  float* erow = E1 + (size_t)row * NE;
  for (int j = lane; j < D_IN; j += 32) erow[j] = s * wrow[j] * d0[j];
  if (lane == 0) { erow[D_IN] = tc; erow[D_IN + 1] = vn; }
  for (int j = D_IN + 2 + lane; j < NE; j += 32) erow[j] = 0.f;
}

// ---------------- Kernel C: C = W2[4096x4096] @ E1[4096x1152] via f32 WMMA ----------------
// WG = 256 threads (8 waves). WG tile 128x128. Wave tile 32x64 (2x4 of 16x16).
// LDS double-buffered; A filled by async b128, B filled transposed by async b32,
// so WMMA fragments are single contiguous ds_load_b64 with no VGPR repacking.
__global__ __launch_bounds__(256) void wmma_gemm_f32(
    const float* __restrict__ A,      // W2, row-major [HID x HID]
    const float* __restrict__ B,      // E1, row-major [HID x NE]
    float* __restrict__ C)            // [HID x NE]
{
  __shared__ float ldsA[2][128 * 16];   // [m][k] row-major
  __shared__ float ldsB[2][128 * 16];   // transposed: [n][k]

  const int K = HID;
  int tid   = threadIdx.x;
  int wave  = tid >> 5;
  int lane  = tid & 31;
  int waveM = wave & 3;               // 0..3  -> 32-row slab
  int waveN = wave >> 2;              // 0..1  -> 64-col slab
  int mBase = blockIdx.y * 128;
  int nBase = blockIdx.x * 128;

  v8f zero = {0.f, 0.f, 0.f, 0.f, 0.f, 0.f, 0.f, 0.f};
  v8f acc[2][4];
#pragma unroll
  for (int i = 0; i < 2; i++)
#pragma unroll
    for (int j = 0; j < 4; j++) acc[i][j] = zero;

  int mLane = lane & 15;
  int kh    = (lane >> 4) * 2;        // lanes 0-15: K={0,1}; lanes 16-31: K={2,3}

#define PREFETCH_TILE(buf, k0)                                                   \
  do {                                                                           \
    _Pragma("unroll")                                                            \
    for (int i = 0; i < 2; i++) {                                                \
      int f = tid + i * 256;          /* float4 index 0..511 */                  \
      int row = f >> 2, c4 = (f & 3) * 4;                                        \
      async_b128(lds_off(&ldsA[buf][row * 16 + c4]),                             \
                 A + (size_t)(mBase + row) * K + (k0) + c4);                     \
    }                                                                            \
    _Pragma("unroll")                                                            \
    for (int i = 0; i < 8; i++) {                                                \
      int e = tid + i * 256;          /* element index 0..2047 */                \
      int kr = e >> 7, n = e & 127;   /* global [k][n] -> LDS [n][k] */          \
      async_b32(lds_off(&ldsB[buf][n * 16 + kr]),                                \
                B + (size_t)((k0) + kr) * NE + nBase + n);                       \
    }                                                                            \
  } while (0)

  PREFETCH_TILE(0, 0);

  for (int k0 = 0; k0 < K; k0 += 16) {
    int cur = (k0 >> 4) & 1;
    wait_async0();                    // this wave's copies into ldsX[cur] done
    __syncthreads();                  // everyone's copies visible WGP-wide
    if (k0 + 16 < K) PREFETCH_TILE(cur ^ 1, k0 + 16);

#pragma unroll
    for (int kk = 0; kk < 4; kk++) {
      int kb = kk * 4 + kh;
      v2f a[2], b[4];
#pragma unroll
      for (int mt = 0; mt < 2; mt++)
        a[mt] = *(const v2f*)&ldsA[cur][(waveM * 32 + mt * 16 + mLane) * 16 + kb];
#pragma unroll
      for (int nt = 0; nt < 4; nt++)
        b[nt] = *(const v2f*)&ldsB[cur][(waveN * 64 + nt * 16 + mLane) * 16 + kb];
#pragma unroll
      for (int mt = 0; mt < 2; mt++)
#pragma unroll
        for (int nt = 0; nt < 4; nt++)
          acc[mt][nt] = __builtin_amdgcn_wmma_f32_16x16x4_f32(
              false, a[mt], false, b[nt], (short)0, acc[mt][nt], false, false);
    }
    __syncthreads();                  // drains DScnt: all reads of ldsX[cur] done
  }
#undef PREFETCH_TILE

  // C/D layout: VGPR r: lanes 0-15 -> M = r, N = lane; lanes 16-31 -> M = r+8, N = lane-16
  int nLane = lane & 15;
  int mHalf = (lane >> 4) * 8;
#pragma unroll
  for (int mt = 0; mt < 2; mt++)
#pragma unroll
    for (int nt = 0; nt < 4; nt++) {
      int col = nBase + waveN * 64 + nt * 16 + nLane;
#pragma unroll
      for (int r = 0; r < 8; r++) {
        int rowg = mBase + waveM * 32 + mt * 16 + mHalf + r;
        C[(size_t)rowg * NE + col] = acc[mt][nt][r];
      }
    }
}

// ---------------- Kernel D: second relu-zono, build E2 ----------------
__global__ __launch_bounds__(256) void relu2_build(
    const float* __restrict__ C2, const float* __restrict__ b2,
    float* __restrict__ E2) {
  int wave = threadIdx.x >> 5;
  int lane = threadIdx.x & 31;
  int row  = blockIdx.x * 8 + wave;

  const float* crow = C2 + (size_t)row * NE;
  float v2 = crow[1025] + b2[row];
  float dr = 0.f;
  for (int j = lane; j < 1025; j += 32) dr += fabsf(crow[j]);
  for (int off = 16; off; off >>= 1) dr += __shfl_xor(dr, off, 32);

  float r = dr;
  float u = v2 + r, l = v2 - r;
  float denom = u - l;
  float slope = u / ((denom == 0.f) ? 1.f : denom);
  float term  = (1.f - slope) * u * 0.5f;
  bool crossing = (u > 0.f) && (l < 0.f);
  bool dead     = (u <= 0.f);
  float s  = dead ? 0.f : (crossing ? slope : 1.f);
  float tc = dead ? 0.f : (crossing ? term  : 0.f);
  float vn = dead ? 0.f : (crossing ? (slope * v2 + term) : v2);

  float* erow = E2 + (size_t)row * NE;
  for (int j = lane; j < 1025; j += 32) erow[j] = s * crow[j];
  if (lane == 0) { erow[1025] = tc; erow[1026] = vn; }
  for (int j = 1027 + lane; j < NE; j += 32) erow[j] = 0.f;
}

// ---------------- Kernel E: final linear + bound extraction ----------------
__global__ __launch_bounds__(256) void final_out(
    const float* __restrict__ W3, const float* __restrict__ b3,
    const float* __restrict__ E2, float* __restrict__ out) {
  __shared__ float red[256];
  __shared__ float vshared;
  int o   = blockIdx.x;
  int tid = threadIdx.x;
  const float* wrow = W3 + (size_t)o * HID;

  float local_abs = 0.f;
  for (int j = tid; j <= 1026; j += 256) {
    float dot = 0.f;
    for (int k = 0; k < HID; k++) dot += wrow[k] * E2[(size_t)k * NE + j];
    if (j < 1026) local_abs += fabsf(dot);
    else          vshared = dot;                 // j == 1026 (value column)
  }
  red[tid] = local_abs;
  __syncthreads();
  for (int s = 128; s; s >>= 1) {
    if (tid < s) red[tid] += red[tid + s];
    __syncthreads();
  }
  if (tid == 0) {
    float v3 = vshared + b3[o];
    float r  = red[0];
    out[o]         = v3 + r;   // upper
    out[O_OUT + o] = v3 - r;   // lower
  }
}

extern "C" void kernel_launch(void* const* d_in, const int* in_sizes, int n_in,
                              void* d_out, int out_size, void* d_ws, size_t ws_size,
                              hipStream_t stream) {
  const float* inp = (const float*)d_in[0];
  const float* W1  = (const float*)d_in[1];
  const float* b1  = (const float*)d_in[2];
  const float* W2  = (const float*)d_in[3];
  const float* b2  = (const float*)d_in[4];
  const float* W3  = (const float*)d_in[5];
  const float* b3  = (const float*)d_in[6];
  float* out = (float*)d_out;

  float* v0   = (float*)d_ws;                    // 1024
  float* dvec = v0 + D_IN;                       // 1024
  float* buf0 = dvec + D_IN;                     // HID*NE : E1, later reused as E2
  float* buf1 = buf0 + (size_t)HID * NE;         // HID*NE : GEMM output

  zono_init<<<4, 256, 0, stream>>>(inp, v0, dvec);
  layer1_relu<<<HID / 8, 256, 0, stream>>>(W1, b1, v0, dvec, buf0);
  dim3 grid(NE / 128, HID / 128);                // 9 x 32 workgroups
  wmma_gemm_f32<<<grid, 256, 0, stream>>>(W2, buf0, buf1);
  relu2_build<<<HID / 8, 256, 0, stream>>>(buf1, b2, buf0);
  final_out<<<O_OUT, 256, 0, stream>>>(W3, b3, buf0, out);
}